// GraphConvBlock_47321949667549
// MI455X (gfx1250) — compile-verified
//
#include <hip/hip_runtime.h>

#define N_NODES 50000
#define N_EDGES 640000
#define IN_CH 128
#define OUT_CH 128
#define EPS_BN 1e-5f
#define NEG_SLOPE 0.01f

typedef __attribute__((ext_vector_type(2))) float v2f;
typedef __attribute__((ext_vector_type(8))) float v8f;

__device__ __forceinline__ void atomicAddF(float* p, float v) {
    // lowers to global_atomic_add_f32 (no CAS loop) on gfx1250
    unsafeAtomicAdd(p, v);
}

__device__ __forceinline__ float leaky(float v) {
    return v >= 0.0f ? v : NEG_SLOPE * v;
}

// ---------------------------------------------------------------------------
// K0: init — zero the output accumulator + channel sums, deg = 1 (self loop)
// ---------------------------------------------------------------------------
__global__ __launch_bounds__(256) void k_init(float* __restrict__ out,
                                              float* __restrict__ deg,
                                              float* __restrict__ csum,
                                              float* __restrict__ csumsq) {
    int i = blockIdx.x * blockDim.x + threadIdx.x;
    if (i < N_NODES * OUT_CH) out[i] = 0.0f;
    if (i < N_NODES) deg[i] = 1.0f;          // self-loop contribution
    if (i < OUT_CH) { csum[i] = 0.0f; csumsq[i] = 0.0f; }
}

// ---------------------------------------------------------------------------
// K1: xw = x @ W via V_WMMA_F32_16X16X4_F32.
// One wave (32 lanes) per 16-row strip of x; N=128 handled as 8 tiles of 16,
// A-fragment reused across the 8 WMMAs of each K-step. K=128 = 32 steps of 4.
//
// ISA layouts (wave32):
//   A (16x4 f32):  lane = {m=lane&15, kh=lane>>4}; v0=A[m][2kh], v1=A[m][2kh+1]
//   B (4x16 f32):  lane = {n=lane&15, kh=lane>>4}; v0=B[2kh][n], v1=B[2kh+1][n]
//   C/D (16x16):   vgpr i -> row i + 8*(lane>>4), col lane&15
// ---------------------------------------------------------------------------
__global__ __launch_bounds__(32) void k_gemm(const float* __restrict__ x,
                                             const float* __restrict__ W,
                                             float* __restrict__ xw) {
    const int lane = threadIdx.x;
    const int m  = lane & 15;
    const int kh = lane >> 4;
    const int rowBase = blockIdx.x * 16;

    v8f acc[8];
    const v8f zero = {0.f, 0.f, 0.f, 0.f, 0.f, 0.f, 0.f, 0.f};
#pragma unroll
    for (int nt = 0; nt < 8; ++nt) acc[nt] = zero;

    const float* xrow = x + (size_t)(rowBase + m) * IN_CH;

    for (int k0 = 0; k0 < IN_CH; k0 += 4) {
        const int ka = k0 + kh * 2;
        v2f a;
        a.x = xrow[ka];
        a.y = xrow[ka + 1];
        const float* w0 = W + (size_t)ka * OUT_CH + m;
        const float* w1 = w0 + OUT_CH;
#pragma unroll
        for (int nt = 0; nt < 8; ++nt) {
            v2f bf;
            bf.x = w0[nt * 16];
            bf.y = w1[nt * 16];
            acc[nt] = __builtin_amdgcn_wmma_f32_16x16x4_f32(
                /*neg_a=*/false, a, /*neg_b=*/false, bf,
                /*c_mod=*/(short)0, acc[nt],
                /*reuse_a=*/false, /*reuse_b=*/false);
        }
    }

#pragma unroll
    for (int nt = 0; nt < 8; ++nt) {
#pragma unroll
        for (int i = 0; i < 8; ++i) {
            xw[(size_t)(rowBase + i + 8 * kh) * OUT_CH + nt * 16 + m] = acc[nt][i];
        }
    }
}

// ---------------------------------------------------------------------------
// K2: deg[dst] += 1 over edges (deg pre-initialized to 1 for self loops)
// ---------------------------------------------------------------------------
__global__ __launch_bounds__(256) void k_deg(const int* __restrict__ ei,
                                             float* __restrict__ deg) {
    int e = blockIdx.x * blockDim.x + threadIdx.x;
    if (e < N_EDGES) {
        int d = ei[N_EDGES + e];  // edge_index[1][e]
        atomicAddF(&deg[d], 1.0f);
    }
}

// ---------------------------------------------------------------------------
// K3: dis[i] = rsqrt(deg[i])   (deg >= 1 always, matches reference where()
// ---------------------------------------------------------------------------
__global__ __launch_bounds__(256) void k_dis(const float* __restrict__ deg,
                                             float* __restrict__ dis) {
    int i = blockIdx.x * blockDim.x + threadIdx.x;
    if (i < N_NODES) dis[i] = rsqrtf(deg[i]);
}

// ---------------------------------------------------------------------------
// K4: edge scatter. One 32-lane wave per edge, float4 per lane (128 channels).
// out[dst,:] += xw[src,:] * dis[src]*dis[dst]  via hw f32 atomics.
// xw (25.6MB) and out (25.6MB) both resident in the 192MB L2.
// ---------------------------------------------------------------------------
__global__ __launch_bounds__(256) void k_scatter(const int* __restrict__ ei,
                                                 const float* __restrict__ xw,
                                                 const float* __restrict__ dis,
                                                 float* __restrict__ out) {
    const int lane = threadIdx.x & 31;
    const int e = blockIdx.x * 8 + (threadIdx.x >> 5);
    if (e >= N_EDGES) return;
    const int s = ei[e];
    const int d = ei[N_EDGES + e];
    const float nrm = dis[s] * dis[d];
    const float4 v = ((const float4*)(xw + (size_t)s * OUT_CH))[lane];
    float* o = out + (size_t)d * OUT_CH + lane * 4;
    atomicAddF(o + 0, v.x * nrm);
    atomicAddF(o + 1, v.y * nrm);
    atomicAddF(o + 2, v.z * nrm);
    atomicAddF(o + 3, v.w * nrm);
}

// ---------------------------------------------------------------------------
// K5: h = leaky(agg + xw*dis^2 + b), written in place to out; per-channel
// sum/sumsq accumulated in registers (fixed channel quad per thread),
// reduced through LDS, then 256 global atomics per block.
// tid = {q = tid&31 channel-quad, nl = tid>>5 node-lane (8 nodes/block/iter)}
// ---------------------------------------------------------------------------
__global__ __launch_bounds__(256) void k_post(const float* __restrict__ xw,
                                              const float* __restrict__ dis,
                                              const float* __restrict__ b,
                                              float* __restrict__ out,
                                              float* __restrict__ csum,
                                              float* __restrict__ csumsq) {
    __shared__ float4 sSum[256];
    __shared__ float4 sSq[256];

    const int tid = threadIdx.x;
    const int q  = tid & 31;   // channel quad: channels 4q..4q+3
    const int nl = tid >> 5;   // 0..7

    const float4 bb = ((const float4*)b)[q];
    float4 ls = {0.f, 0.f, 0.f, 0.f};
    float4 lq = {0.f, 0.f, 0.f, 0.f};

    for (int node = blockIdx.x * 8 + nl; node < N_NODES; node += gridDim.x * 8) {
        const float dd = dis[node];
        const float w  = dd * dd;                    // self-loop norm
        float4 a  = ((float4*)(out + (size_t)node * OUT_CH))[q];
        float4 xv = ((const float4*)(xw + (size_t)node * OUT_CH))[q];
        float4 h;
        h.x = leaky(a.x + xv.x * w + bb.x);
        h.y = leaky(a.y + xv.y * w + bb.y);
        h.z = leaky(a.z + xv.z * w + bb.z);
        h.w = leaky(a.w + xv.w * w + bb.w);
        ((float4*)(out + (size_t)node * OUT_CH))[q] = h;
        ls.x += h.x; ls.y += h.y; ls.z += h.z; ls.w += h.w;
        lq.x += h.x * h.x; lq.y += h.y * h.y; lq.z += h.z * h.z; lq.w += h.w * h.w;
    }

    sSum[tid] = ls;
    sSq[tid]  = lq;
    __syncthreads();

    if (nl == 0) {
        float4 ts = {0.f, 0.f, 0.f, 0.f};
        float4 tq = {0.f, 0.f, 0.f, 0.f};
        for (int j = 0; j < 8; ++j) {
            float4 a = sSum[j * 32 + q];
            float4 c = sSq[j * 32 + q];
            ts.x += a.x; ts.y += a.y; ts.z += a.z; ts.w += a.w;
            tq.x += c.x; tq.y += c.y; tq.z += c.z; tq.w += c.w;
        }
        atomicAddF(&csum[q * 4 + 0], ts.x);
        atomicAddF(&csum[q * 4 + 1], ts.y);
        atomicAddF(&csum[q * 4 + 2], ts.z);
        atomicAddF(&csum[q * 4 + 3], ts.w);
        atomicAddF(&csumsq[q * 4 + 0], tq.x);
        atomicAddF(&csumsq[q * 4 + 1], tq.y);
        atomicAddF(&csumsq[q * 4 + 2], tq.z);
        atomicAddF(&csumsq[q * 4 + 3], tq.w);
    }
}

// ---------------------------------------------------------------------------
// K6: out = gamma * (h - mean) * rsqrt(var + eps) + beta  (biased var)
// ---------------------------------------------------------------------------
__global__ __launch_bounds__(256) void k_norm(const float* __restrict__ csum,
                                              const float* __restrict__ csumsq,
                                              const float* __restrict__ gamma,
                                              const float* __restrict__ beta,
                                              float* __restrict__ out) {
    const int i = blockIdx.x * blockDim.x + threadIdx.x;  // quad index
    if (i >= N_NODES * (OUT_CH / 4)) return;
    const int q = i & 31;
    const float invN = 1.0f / (float)N_NODES;
    const float4 s  = ((const float4*)csum)[q];
    const float4 sq = ((const float4*)csumsq)[q];
    const float4 g  = ((const float4*)gamma)[q];
    const float4 bt = ((const float4*)beta)[q];
    float4 h = ((float4*)out)[i];

    float mx = s.x * invN, my = s.y * invN, mz = s.z * invN, mw = s.w * invN;
    float vx = sq.x * invN - mx * mx;
    float vy = sq.y * invN - my * my;
    float vz = sq.z * invN - mz * mz;
    float vw = sq.w * invN - mw * mw;

    h.x = g.x * (h.x - mx) * rsqrtf(vx + EPS_BN) + bt.x;
    h.y = g.y * (h.y - my) * rsqrtf(vy + EPS_BN) + bt.y;
    h.z = g.z * (h.z - mz) * rsqrtf(vz + EPS_BN) + bt.z;
    h.w = g.w * (h.w - mw) * rsqrtf(vw + EPS_BN) + bt.w;

    ((float4*)out)[i] = h;
}

// ---------------------------------------------------------------------------
extern "C" void kernel_launch(void* const* d_in, const int* in_sizes, int n_in,
                              void* d_out, int out_size, void* d_ws, size_t ws_size,
                              hipStream_t stream) {
    const float* x     = (const float*)d_in[0];
    const int*   ei    = (const int*)d_in[1];   // [2, N_EDGES] flattened
    const float* W     = (const float*)d_in[2];
    const float* b     = (const float*)d_in[3];
    const float* gamma = (const float*)d_in[4];
    const float* beta  = (const float*)d_in[5];
    float* out = (float*)d_out;

    // workspace layout (floats): xw | deg | dis | csum | csumsq  (~26 MB)
    float* xw     = (float*)d_ws;
    float* deg    = xw + (size_t)N_NODES * OUT_CH;
    float* dis    = deg + N_NODES;
    float* csum   = dis + N_NODES;
    float* csumsq = csum + OUT_CH;

    k_init<<<(N_NODES * OUT_CH + 255) / 256, 256, 0, stream>>>(out, deg, csum, csumsq);
    k_gemm<<<N_NODES / 16, 32, 0, stream>>>(x, W, xw);          // 50000/16 = 3125
    k_deg<<<(N_EDGES + 255) / 256, 256, 0, stream>>>(ei, deg);
    k_dis<<<(N_NODES + 255) / 256, 256, 0, stream>>>(deg, dis);
    k_scatter<<<N_EDGES / 8, 256, 0, stream>>>(ei, xw, dis, out);
    k_post<<<256, 256, 0, stream>>>(xw, dis, b, out, csum, csumsq);
    k_norm<<<(N_NODES * (OUT_CH / 4) + 255) / 256, 256, 0, stream>>>(csum, csumsq, gamma, beta, out);
}